// TokenEncoder_22101901705941
// MI455X (gfx1250) — compile-verified
//
#include <hip/hip_runtime.h>
#include <stdint.h>

#define DIM      512
#define MAX_POS  2048
#define NTOK     (8 * 16 * 128)
#define SWEEPS   10        // max path depth for MAX_POS=2048
#define KCHUNK   64        // K staging chunk (double-buffered in LDS)

typedef float v2f __attribute__((ext_vector_type(2)));
typedef float v8f __attribute__((ext_vector_type(8)));

// ---------------------------------------------------------------------------
// gfx1250 async global->LDS copy (ASYNCcnt-tracked), via inline asm so it is
// portable across ROCm-7.2 / amdgpu-toolchain builtin-arity differences.
// VDST VGPR carries the LDS byte offset (ISA 10.2: LDS flat addresses map by
// discarding the upper bits, so the low 32 bits of a generic pointer to
// __shared__ are the LDS offset).
// ---------------------------------------------------------------------------
__device__ __forceinline__ void async_g2l_b128(const float* gsrc, float* lds_dst) {
    uint32_t loff = (uint32_t)(uintptr_t)lds_dst;
    asm volatile("global_load_async_to_lds_b128 %0, %1, off"
                 :: "v"(loff), "v"(gsrc)
                 : "memory");
}

__device__ __forceinline__ void async_wait_all() {
    asm volatile("s_wait_asynccnt 0x0" ::: "memory");
}

// ---------------------------------------------------------------------------
// Kernel 0: maps = ones(2048, 512)
// ---------------------------------------------------------------------------
__global__ void init_ones_kernel(float* __restrict__ buf, int n) {
    int i = blockIdx.x * blockDim.x + threadIdx.x;
    if (i < n) buf[i] = 1.0f;
}

// ---------------------------------------------------------------------------
// Kernel 1: one sweep of the path-map recurrence.
//   m0 = src @ P0^T ; m1 = src @ P1^T
//   dst[p] = (p >= 2^(d+1)) ? ((p>>d)&1 ? m1[p] : m0[p]) : src[p]
// Grid: x = 16 N-blocks (32 cols), y = 16 M-groups (8 waves x 16 rows).
// Each wave32 owns a 16x32 output footprint for BOTH matrices:
// 4 f32 WMMA accumulators, so each A-fragment load feeds 4 WMMAs.
// P0/P1 panels are double-buffered in LDS via async global->LDS copies.
// ---------------------------------------------------------------------------
__launch_bounds__(256)
__global__ void path_sweep_kernel(const float* __restrict__ src,
                                  float*       __restrict__ dst,
                                  const float* __restrict__ P0,
                                  const float* __restrict__ P1,
                                  int d) {
    // [buffer][matrix][row in 32-col panel][68-float padded K-chunk]
    // 68 mod 64 == 4  =>  b64 B-fragment reads hit 64 distinct banks.
    __shared__ float sP[2][2][32][68];

    const int tid   = threadIdx.x;
    const int lane  = tid & 31;        // wave32 lane
    const int wave  = tid >> 5;        // 8 waves per block
    const int n0    = blockIdx.x * 32;
    const int pbase = (blockIdx.y * 8 + wave) * 16;
    const int r     = lane & 15;
    const int koff  = (lane < 16) ? 0 : 2;   // ISA 32-bit A/B fragment layout

    v8f acc00 = {};   // src @ P0^T, N-tile 0
    v8f acc01 = {};   // src @ P0^T, N-tile 1
    v8f acc10 = {};   // src @ P1^T, N-tile 0
    v8f acc11 = {};   // src @ P1^T, N-tile 1

    const float* Pm[2] = {P0, P1};

    // ---- stage one K-chunk of both 32-row panels into LDS buffer `buf` ----
    // 2 matrices x 32 rows x 16 float4 = 1024 b128 copies / 256 threads = 4.
    auto stage = [&](int kc, int buf) {
        #pragma unroll
        for (int it = 0; it < 4; ++it) {
            int i  = it * 256 + tid;
            int m  = i >> 9;           // matrix 0/1
            int nn = (i >> 4) & 31;    // panel row 0..31
            int k4 = i & 15;           // float4 within K-chunk
            async_g2l_b128(&Pm[m][(n0 + nn) * 512 + kc + k4 * 4],
                           &sP[buf][m][nn][k4 * 4]);
        }
    };

    stage(0, 0);
    async_wait_all();
    __syncthreads();

    for (int c = 0; c < DIM / KCHUNK; ++c) {
        const int buf = c & 1;
        if (c + 1 < DIM / KCHUNK) stage((c + 1) * KCHUNK, buf ^ 1);

        const float* arow = &src[(pbase + r) * 512 + c * KCHUNK + koff];
        #pragma unroll 4
        for (int k = 0; k < KCHUNK; k += 4) {
            // A fragment 16x4: lanes 0-15 hold K=k..k+1, lanes 16-31 K=k+2..k+3
            v2f a   = *(const v2f*)(arow + k);
            // B fragments 4x16 (= P^T tiles): element (kk, n) = P[n, kk]
            v2f b00 = *(const v2f*)&sP[buf][0][r][k + koff];
            v2f b01 = *(const v2f*)&sP[buf][0][r + 16][k + koff];
            v2f b10 = *(const v2f*)&sP[buf][1][r][k + koff];
            v2f b11 = *(const v2f*)&sP[buf][1][r + 16][k + koff];
            acc00 = __builtin_amdgcn_wmma_f32_16x16x4_f32(
                        false, a, false, b00, (short)0, acc00, false, false);
            acc01 = __builtin_amdgcn_wmma_f32_16x16x4_f32(
                        false, a, false, b01, (short)0, acc01, false, false);
            acc10 = __builtin_amdgcn_wmma_f32_16x16x4_f32(
                        false, a, false, b10, (short)0, acc10, false, false);
            acc11 = __builtin_amdgcn_wmma_f32_16x16x4_f32(
                        false, a, false, b11, (short)0, acc11, false, false);
        }

        async_wait_all();     // next chunk's panels landed in LDS
        __syncthreads();      // everyone done reading this chunk's buffer
    }

    // Epilogue: per-row bit select (C/D layout: VGPR v -> M = v (+8 for lanes>=16))
    const int      na     = n0 + (lane & 15);
    const int      nb     = na + 16;
    const int      mhalf  = (lane >> 4) << 3;
    const unsigned thresh = 2u << d;
    #pragma unroll
    for (int v = 0; v < 8; ++v) {
        int p = pbase + mhalf + v;
        float va, vb;
        if ((unsigned)p >= thresh) {
            int bit = (p >> d) & 1;
            va = bit ? acc10[v] : acc00[v];
            vb = bit ? acc11[v] : acc01[v];
        } else {
            va = src[p * 512 + na];
            vb = src[p * 512 + nb];
        }
        dst[p * 512 + na] = va;
        dst[p * 512 + nb] = vb;
    }
}

// ---------------------------------------------------------------------------
// Kernel 2: gather + fused 4->8->1 SiLU MLP per (token, dim).
// One thread handles one token and 4 consecutive dims (float4 path).
// ---------------------------------------------------------------------------
__launch_bounds__(256)
__global__ void encode_kernel(const int*   __restrict__ token_types,
                              const int*   __restrict__ token_values,
                              const int*   __restrict__ node_positions,
                              const int*   __restrict__ tree_positions,
                              const float* __restrict__ type_emb,
                              const float* __restrict__ op_emb,
                              const float* __restrict__ leaf_emb,
                              const float* __restrict__ ref_emb,
                              const float* __restrict__ db_pe,
                              const float* __restrict__ w1,
                              const float* __restrict__ b1,
                              const float* __restrict__ w2,
                              const float* __restrict__ b2,
                              const float* __restrict__ maps,
                              float*       __restrict__ out) {
    __shared__ float sw1[32];
    __shared__ float sb1[8];
    __shared__ float sw2[8];
    __shared__ float sb2;
    if (threadIdx.x < 32) sw1[threadIdx.x] = w1[threadIdx.x];
    if (threadIdx.x < 8) {
        sb1[threadIdx.x] = b1[threadIdx.x];
        sw2[threadIdx.x] = w2[threadIdx.x];
    }
    if (threadIdx.x == 0) sb2 = b2[0];
    __syncthreads();

    const int gid   = blockIdx.x * 256 + threadIdx.x;
    const int tok   = gid >> 7;          // 128 dim-groups per token
    const int dbase = (gid & 127) * 4;

    const int tt = token_types[tok];
    const int tv = token_values[tok];
    const int np = node_positions[tok];
    const int tp = tree_positions[tok];

    const float* tvbase = (tt == 0) ? op_emb
                        : (tt == 1) ? leaf_emb
                        : (tt == 2) ? ref_emb
                        : db_pe;
    const int gidx = (tp == -1) ? 1001 : (tp < 0 ? 0 : tp);

    const float4 te = *(const float4*)&type_emb[tt   * 512 + dbase];
    const float4 ve = *(const float4*)&tvbase  [tv   * 512 + dbase];
    const float4 pe = *(const float4*)&maps    [np   * 512 + dbase];
    const float4 ge = *(const float4*)&ref_emb [gidx * 512 + dbase];

    const float c0a[4] = {te.x, te.y, te.z, te.w};
    const float c1a[4] = {ve.x, ve.y, ve.z, ve.w};
    const float c2a[4] = {pe.x, pe.y, pe.z, pe.w};
    const float c3a[4] = {ge.x, ge.y, ge.z, ge.w};

    float res[4];
    #pragma unroll
    for (int j = 0; j < 4; ++j) {
        const float c0 = c0a[j], c1 = c1a[j], c2 = c2a[j], c3 = c3a[j];
        float acc = sb2;
        #pragma unroll
        for (int o = 0; o < 8; ++o) {
            float z = sw1[o * 4 + 0] * c0 + sw1[o * 4 + 1] * c1 +
                      sw1[o * 4 + 2] * c2 + sw1[o * 4 + 3] * c3 + sb1[o];
            float h = z * (1.0f / (1.0f + __expf(-z)));   // SiLU
            acc += h * sw2[o];
        }
        res[j] = acc;
    }
    float4 ov = {res[0], res[1], res[2], res[3]};
    *(float4*)&out[tok * 512 + dbase] = ov;
}

// ---------------------------------------------------------------------------
extern "C" void kernel_launch(void* const* d_in, const int* in_sizes, int n_in,
                              void* d_out, int out_size, void* d_ws, size_t ws_size,
                              hipStream_t stream) {
    (void)in_sizes; (void)n_in; (void)out_size; (void)ws_size;

    const int*   token_types    = (const int*)d_in[0];
    const int*   token_values   = (const int*)d_in[1];
    const int*   node_positions = (const int*)d_in[2];
    const int*   tree_positions = (const int*)d_in[3];
    const float* type_emb = (const float*)d_in[4];
    const float* op_emb   = (const float*)d_in[5];
    const float* leaf_emb = (const float*)d_in[6];
    const float* ref_emb  = (const float*)d_in[7];
    const float* db_pe    = (const float*)d_in[8];
    const float* prims    = (const float*)d_in[9];
    const float* w1       = (const float*)d_in[10];
    const float* b1       = (const float*)d_in[11];
    const float* w2       = (const float*)d_in[12];
    const float* b2       = (const float*)d_in[13];

    float* bufA = (float*)d_ws;                 // 2048*512 f32 = 4 MB
    float* bufB = bufA + MAX_POS * DIM;         // ping-pong partner
    const float* P0 = prims;                    // primitives[0], 512x512
    const float* P1 = prims + DIM * DIM;        // primitives[1]

    init_ones_kernel<<<(MAX_POS * DIM) / 256, 256, 0, stream>>>(bufA, MAX_POS * DIM);

    float* s = bufA;
    float* t = bufB;
    for (int d = 0; d < SWEEPS; ++d) {
        path_sweep_kernel<<<dim3(16, 16), 256, 0, stream>>>(s, t, P0, P1, d);
        float* tmp = s; s = t; t = tmp;
    }
    // SWEEPS == 10 (even) -> final table back in bufA (== s)

    encode_kernel<<<(NTOK * 128) / 256, 256, 0, stream>>>(
        token_types, token_values, node_positions, tree_positions,
        type_emb, op_emb, leaf_emb, ref_emb, db_pe,
        w1, b1, w2, b2, s, (float*)d_out);
}